// BandSplit_57320633532813
// MI455X (gfx1250) — compile-verified
//
#include <hip/hip_runtime.h>

typedef float v2f __attribute__((ext_vector_type(2)));
typedef float v8f __attribute__((ext_vector_type(8)));

#define NF     1025
#define NBANDS 257
#define OCH    128
#define MPOS   2048   // B*C*T = 4*2*256
#define MTILE  64     // M positions per block

// Closed-form band geometry (all mel weights are 1.0 in the reference pattern).
__device__ __forceinline__ void band_geom(int f, int& base, int& width) {
  if (f < 128)      { base = f;               width = 1; }
  else if (f < 160) { base = 128 + (f-128)*4; width = 4; }
  else if (f < 192) { base = 256 + (f-160)*8; width = 8; }
  else if (f < 256) { base = 512 + (f-192)*8; width = 8; }
  else              { base = 1024;            width = 1; }
}

// Pack pre_w into wp[f][w][o][i] = pre_w[i][base(f)+w][o], zero for w >= width.
// Layout chosen so a B-matrix fragment lane load is one contiguous b64 (K-pair).
__global__ void pack_weights_kernel(const float* __restrict__ pre_w,
                                    float* __restrict__ wp) {
  int idx = blockIdx.x * blockDim.x + threadIdx.x;
  if (idx >= NBANDS * 8 * OCH * 2) return;
  int i = idx & 1;
  int o = (idx >> 1) & (OCH - 1);
  int w = (idx >> 8) & 7;
  int f = idx >> 11;
  int base, width;
  band_geom(f, base, width);
  float v = 0.0f;
  if (w < width) {
    int bin = base + w;
    v = pre_w[((size_t)i * NF + bin) * OCH + o];
  }
  wp[idx] = v;
}

// One wave: 16 output channels (nbase..nbase+15) x 64 positions (4 C-tiles),
// accumulated with V_WMMA_F32_16X16X4_F32 over NSTEPS k-steps of 4.
template <int NSTEPS>
__device__ __forceinline__ void band_tile(const float* __restrict__ x,
                                          const float* __restrict__ wpf,
                                          float bias, float* __restrict__ out,
                                          int f, int mbase, int base,
                                          int nbase, int lane) {
  const int lhalf = lane >> 4;   // 0: K pair (4s,4s+1)   1: K pair (4s+2,4s+3)
  const int l15   = lane & 15;

  // B fragments: wp[f][kp][o][0:2] is the contiguous (K=2kp, 2kp+1) pair.
  v2f bfrag[NSTEPS];
#pragma unroll
  for (int s = 0; s < NSTEPS; ++s) {
    int kp = 2 * s + lhalf;
    bfrag[s] = *(const v2f*)(wpf + ((size_t)kp * OCH + nbase + l15) * 2);
  }

#pragma unroll
  for (int m = 0; m < 4; ++m) {
    int pos = mbase + m * 16 + l15;          // A rows: M = l15 in both halves
    v8f acc = {bias, bias, bias, bias, bias, bias, bias, bias};
#pragma unroll
    for (int s = 0; s < NSTEPS; ++s) {
      int bin = base + 2 * s + lhalf;        // w index; K = 2w + i, i contiguous
      bin = bin > (NF - 1) ? (NF - 1) : bin; // clamp (masked by zero B rows)
      v2f afrag = *(const v2f*)(x + ((size_t)pos * NF + bin) * 2);
      acc = __builtin_amdgcn_wmma_f32_16x16x4_f32(
          false, afrag, false, bfrag[s], (short)0, acc, false, false);
    }
    // D layout: VGPR j -> M = j + 8*lhalf, N = l15 (per ISA 16x16 f32 C/D map)
    int mrow = mbase + m * 16 + lhalf * 8;
#pragma unroll
    for (int j = 0; j < 8; ++j) {
      size_t off = ((size_t)(mrow + j) * NBANDS + f) * OCH + nbase + l15;
      __builtin_nontemporal_store(acc[j], out + off);   // 270 MB streamed once
    }
  }
}

__global__ void bandsplit_wmma_kernel(const float* __restrict__ x,
                                      const float* __restrict__ wp,
                                      const float* __restrict__ pre_b,
                                      float* __restrict__ out) {
  const int f     = blockIdx.y;
  const int mbase = blockIdx.x * MTILE;
  const int wave  = threadIdx.x >> 5;   // 8 waves per block
  const int lane  = threadIdx.x & 31;
  const int nbase = wave * 16;

  int base, width;
  band_geom(f, base, width);
  const float* wpf  = wp + (size_t)f * 8 * OCH * 2;
  const float  bias = pre_b[(size_t)f * OCH + nbase + (lane & 15)];

  // Uniform per block -> EXEC stays all-1s inside each specialization.
  if (width == 1)      band_tile<1>(x, wpf, bias, out, f, mbase, base, nbase, lane);
  else if (width == 4) band_tile<2>(x, wpf, bias, out, f, mbase, base, nbase, lane);
  else                 band_tile<4>(x, wpf, bias, out, f, mbase, base, nbase, lane);
}

extern "C" void kernel_launch(void* const* d_in, const int* in_sizes, int n_in,
                              void* d_out, int out_size, void* d_ws, size_t ws_size,
                              hipStream_t stream) {
  const float* x     = (const float*)d_in[0];   // (4,2,256,1025,2) f32
  const float* pre_w = (const float*)d_in[1];   // (2,1025,128) f32
  const float* pre_b = (const float*)d_in[2];   // (257,128) f32
  float* out = (float*)d_out;                   // (4,2,256,257,128) f32
  float* wp  = (float*)d_ws;                    // packed weights: 2.1 MB

  {
    int total = NBANDS * 8 * OCH * 2;
    int blocks = (total + 255) / 256;
    pack_weights_kernel<<<blocks, 256, 0, stream>>>(pre_w, wp);
  }
  {
    dim3 grid(MPOS / MTILE, NBANDS);  // 32 x 257 blocks
    bandsplit_wmma_kernel<<<grid, 256, 0, stream>>>(x, wp, pre_b, out);
  }
}